// QuantumLayer_34634616275058
// MI455X (gfx1250) — compile-verified
//
#include <hip/hip_runtime.h>

typedef __attribute__((ext_vector_type(2))) float v2f;
typedef __attribute__((ext_vector_type(8))) float v8f;

constexpr int BLOCK = 256;   // 8 waves (wave32)
constexpr int ITERS = 4;     // elements per thread
constexpr int EPB   = BLOCK * ITERS;

// out[e*4+w] = sum_j signs[w][j] * | (M @ exp(i*phi(e)))_j |^2
// M = (1/4) * V(params) * P_cnotchain  (real, 16x16)
__global__ __launch_bounds__(BLOCK) void qsim_kernel(const float* __restrict__ inp,
                                                     const float* __restrict__ par,
                                                     float* __restrict__ out, int n)
{
    __shared__ float Mlds[16 * 16];        // Mlds[t*16 + j] = M[j][t]
    __shared__ float Yr[BLOCK * 17];       // per-element phase vectors (padded rows)
    __shared__ float Yi[BLOCK * 17];

    const int tid  = threadIdx.x;
    const int lane = tid & 31;
    const int wv   = tid >> 5;
    const int half = lane >> 4;            // 0: lanes 0-15, 1: lanes 16-31
    const int jc   = lane & 15;

    // ---------- per-block: build real 16x16 matrix M (one column per thread) ----------
    if (tid < 16) {
        // column t of (V * P): start from basis state P e_t = e_{sigma(t)}
        int k = tid;
        k ^= ((k >> 3) & 1) << 2;          // CNOT(0,1)
        k ^= ((k >> 2) & 1) << 1;          // CNOT(1,2)
        k ^= ((k >> 1) & 1);               // CNOT(2,3)
        float f[16];
        #pragma unroll
        for (int j = 0; j < 16; ++j) f[j] = (j == k) ? 1.0f : 0.0f;

        #pragma unroll
        for (int l = 0; l < 2; ++l) {
            #pragma unroll
            for (int w = 0; w < 4; ++w) {  // RY(params[l][w]) on wire w (bit 3-w)
                float sn, cs;
                __sincosf(0.5f * par[l * 4 + w], &sn, &cs);
                const int mask = 1 << (3 - w);
                #pragma unroll
                for (int k0 = 0; k0 < 16; ++k0) {
                    if (!(k0 & mask)) {
                        float a = f[k0], b = f[k0 | mask];
                        f[k0]        = cs * a - sn * b;
                        f[k0 | mask] = sn * a + cs * b;
                    }
                }
            }
            // CNOTs (c,t): (0,1),(1,2),(2,3),(3,0)  -> new[j] = old[j ^ (bit_c(j)?mask_t:0)]
            const int cw[4] = {0, 1, 2, 3};
            const int tw[4] = {1, 2, 3, 0};
            #pragma unroll
            for (int p = 0; p < 4; ++p) {
                float g[16];
                #pragma unroll
                for (int j = 0; j < 16; ++j) {
                    int src = j ^ (((j >> (3 - cw[p])) & 1) << (3 - tw[p]));
                    g[j] = f[src];
                }
                #pragma unroll
                for (int j = 0; j < 16; ++j) f[j] = g[j];
            }
        }
        #pragma unroll
        for (int j = 0; j < 16; ++j) Mlds[tid * 16 + j] = 0.25f * f[j];
    }
    __syncthreads();

    // ---------- preload B operand: B[k][n] = M[n][k], 4 K-steps of 4 ----------
    // layout: v0 = row (4s + 2*half), v1 = row (4s + 2*half + 1), cols = lane&15
    v2f Bm[4];
    #pragma unroll
    for (int s = 0; s < 4; ++s) {
        Bm[s].x = Mlds[(4 * s + 2 * half) * 16 + jc];
        Bm[s].y = Mlds[(4 * s + 2 * half + 1) * 16 + jc];
    }

    const int blockBase = blockIdx.x * EPB;

    for (int it = 0; it < ITERS; ++it) {
        int e     = blockBase + it * BLOCK + tid;
        int esafe = (e < n) ? e : (n - 1);
        const float4 x = reinterpret_cast<const float4*>(inp)[esafe];

        // per-wire half-angle phases; e^{i phi_t} = prod_w (c_w + i*(2b-1)*s_w)
        float cw_[4], sw_[4];
        __sincosf(0.5f * x.x, &sw_[0], &cw_[0]);
        __sincosf(0.5f * x.y, &sw_[1], &cw_[1]);
        __sincosf(0.5f * x.z, &sw_[2], &cw_[2]);
        __sincosf(0.5f * x.w, &sw_[3], &cw_[3]);

        float pr[16], pi[16];
        pr[0] = 1.0f; pi[0] = 0.0f;
        #pragma unroll
        for (int w = 0; w < 4; ++w) {
            const int sz = 1 << w;
            #pragma unroll
            for (int t = sz - 1; t >= 0; --t) {
                float a = pr[t], b = pi[t];
                float c = cw_[w], s = sw_[w];
                pr[2 * t]     = a * c + b * s;   // bit 0: *(c - i s)
                pi[2 * t]     = b * c - a * s;
                pr[2 * t + 1] = a * c - b * s;   // bit 1: *(c + i s)
                pi[2 * t + 1] = b * c + a * s;
            }
        }

        __syncthreads();   // previous iteration's LDS reads are done
        #pragma unroll
        for (int k2 = 0; k2 < 16; ++k2) {
            Yr[tid * 17 + k2] = pr[k2];
            Yi[tid * 17 + k2] = pi[k2];
        }
        __syncthreads();

        // two 16-element M-tiles per wave
        #pragma unroll
        for (int t = 0; t < 2; ++t) {
            v8f cre = {0.f, 0.f, 0.f, 0.f, 0.f, 0.f, 0.f, 0.f};
            v8f cim = {0.f, 0.f, 0.f, 0.f, 0.f, 0.f, 0.f, 0.f};
            const int rowBase = (wv * 32 + t * 16 + jc) * 17;
            #pragma unroll
            for (int s = 0; s < 4; ++s) {
                const int kk = 4 * s + 2 * half;
                v2f ar, ai;
                ar.x = Yr[rowBase + kk];  ar.y = Yr[rowBase + kk + 1];
                ai.x = Yi[rowBase + kk];  ai.y = Yi[rowBase + kk + 1];
                cre = __builtin_amdgcn_wmma_f32_16x16x4_f32(false, ar, false, Bm[s],
                                                            (short)0, cre, false, false);
                cim = __builtin_amdgcn_wmma_f32_16x16x4_f32(false, ai, false, Bm[s],
                                                            (short)0, cim, false, false);
            }

            float p[8];
            #pragma unroll
            for (int r = 0; r < 8; ++r) p[r] = cre[r] * cre[r] + cim[r] * cim[r];

            // Walsh-Hadamard butterfly across 16 lanes: lane m gets sum_j (-1)^{popcnt(m&j)} p_j
            #pragma unroll
            for (int b = 1; b <= 8; b <<= 1) {
                #pragma unroll
                for (int r = 0; r < 8; ++r) {
                    float o = __shfl_xor(p[r], b, 32);
                    p[r] = (lane & b) ? (o - p[r]) : (o + p[r]);
                }
            }

            // wire w <-> lane mask 1<<(3-w): lanes 8,4,2,1 hold wires 0,1,2,3
            if (jc == 1 || jc == 2 || jc == 4 || jc == 8) {
                const int w = (jc == 8) ? 0 : (jc == 4) ? 1 : (jc == 2) ? 2 : 3;
                const int ebase = blockBase + it * BLOCK + wv * 32 + t * 16 + half * 8;
                #pragma unroll
                for (int r = 0; r < 8; ++r) {
                    int ee = ebase + r;
                    if (ee < n) out[ee * 4 + w] = p[r];
                }
            }
        }
    }
}

extern "C" void kernel_launch(void* const* d_in, const int* in_sizes, int n_in,
                              void* d_out, int out_size, void* d_ws, size_t ws_size,
                              hipStream_t stream) {
    const float* inp = (const float*)d_in[0];   // (B,4) float32
    const float* par = (const float*)d_in[1];   // (2,4) float32
    float* out = (float*)d_out;                 // (B,4) float32
    const int n = in_sizes[0] / 4;
    const int blocks = (n + EPB - 1) / EPB;
    qsim_kernel<<<blocks, BLOCK, 0, stream>>>(inp, par, out, n);
}